// SAGPool_20126216749339
// MI455X (gfx1250) — compile-verified
//
#include <hip/hip_runtime.h>
#include <math.h>

typedef __attribute__((ext_vector_type(2))) float v2f;
typedef __attribute__((ext_vector_type(8))) float v8f;

#define TPB 256

// ---------------- degree / normalization ----------------
__global__ void k_fill_f32(float* p, float v, int n) {
  int i = blockIdx.x * blockDim.x + threadIdx.x;
  if (i < n) p[i] = v;
}
__global__ void k_deg_accum(const int* __restrict__ dst, float* deg, int E) {
  int e = blockIdx.x * blockDim.x + threadIdx.x;
  if (e < E) atomicAdd(&deg[dst[e]], 1.0f);
}
__global__ void k_rsqrt(float* p, int n) {
  int i = blockIdx.x * blockDim.x + threadIdx.x;
  if (i < n) p[i] = rsqrtf(p[i]);
}

// ---------------- GEMM: out[N][32] = A[N][K] @ W[K][32], fp32 WMMA ----------------
// One wave -> 16 rows x 32 cols. A tile 16x4 (v2f/lane), B tile 4x16 (v2f/lane),
// two v8f accumulators (cols 0-15 and 16-31). W staged transposed in LDS, padded
// stride 132 dwords so lanes hit distinct banks, 8B-aligned for ds_load_b64.
__global__ void k_gemm32_wmma(const float* __restrict__ A,
                              const float* __restrict__ W,
                              float* __restrict__ out,
                              int nrows, int K) {
  constexpr int KP = 132;                 // 128 max K + pad (keeps 8B alignment)
  __shared__ float sWT[32 * KP];          // 16.9 KB
  int t = threadIdx.x;
  for (int i = t; i < K * 32; i += TPB) {
    int kk = i >> 5, n = i & 31;          // W row-major [K][32]
    sWT[n * KP + kk] = W[i];
  }
  __syncthreads();

  int lane = t & 31;
  int wave = t >> 5;
  int hi = lane >> 4;                     // 0: K pair {k,k+1}; 1: {k+2,k+3}
  int m  = lane & 15;
  int rowBase = blockIdx.x * 128 + wave * 16;
  int rowL = rowBase + m;
  int rowC = rowL < nrows ? rowL : nrows - 1;   // clamp: keep EXEC all-ones
  const float* aRow = A + (size_t)rowC * K;

  v8f acc0 = {}; v8f acc1 = {};
  for (int k = 0; k < K; k += 4) {
    int ka = k + 2 * hi;
    v2f a  = *(const v2f*)(aRow + ka);                  // A[m][ka..ka+1]
    v2f b0 = *(const v2f*)(sWT + m * KP + ka);          // W[ka..ka+1][m]
    v2f b1 = *(const v2f*)(sWT + (m + 16) * KP + ka);   // W[ka..ka+1][m+16]
    acc0 = __builtin_amdgcn_wmma_f32_16x16x4_f32(false, a, false, b0,
                                                 (short)0, acc0, false, false);
    acc1 = __builtin_amdgcn_wmma_f32_16x16x4_f32(false, a, false, b1,
                                                 (short)0, acc1, false, false);
  }
  // D layout: VGPR r holds M = r + 8*hi, N = m
  int mrow = rowBase + hi * 8;
#pragma unroll
  for (int r = 0; r < 8; ++r) {
    int row = mrow + r;
    if (row < nrows) {
      out[(size_t)row * 32 + m]      = acc0[r];
      out[(size_t)row * 32 + 16 + m] = acc1[r];
    }
  }
}

// ---------------- GCN aggregation ----------------
// h[n][c] = dis[n]^2 * hW[n][c] + b[c]   (self-loop term + bias, init)
__global__ void k_selfterm(const float* __restrict__ hW, const float* __restrict__ dis,
                           const float* __restrict__ b, float* __restrict__ h, int N) {
  int i = blockIdx.x * blockDim.x + threadIdx.x;
  if (i >= N * 32) return;
  int n = i >> 5, c = i & 31;
  float d = dis[n];
  h[i] = d * d * hW[i] + b[c];
}
// h[dst] += dis[src]*dis[dst] * hW[src]   (8 threads/edge, float4 each)
__global__ void k_edge_scatter(const int* __restrict__ src, const int* __restrict__ dst,
                               const float* __restrict__ dis, const float* __restrict__ hW,
                               float* __restrict__ h, int E) {
  int tid = blockIdx.x * blockDim.x + threadIdx.x;
  int e = tid >> 3, q = tid & 7;
  if (e >= E) return;
  int s = src[e], d = dst[e];
  float w = dis[s] * dis[d];
  float4 v = *(const float4*)(hW + (size_t)s * 32 + q * 4);
  float* hp = h + (size_t)d * 32 + q * 4;
  atomicAdd(hp + 0, w * v.x);
  atomicAdd(hp + 1, w * v.y);
  atomicAdd(hp + 2, w * v.z);
  atomicAdd(hp + 3, w * v.w);
}
__global__ void k_relu(float* p, int n) {
  int i = blockIdx.x * blockDim.x + threadIdx.x;
  if (i < n) p[i] = fmaxf(p[i], 0.0f);
}

// ---------------- scoring layer (32 -> 1) ----------------
__global__ void k_score_pre(const float* __restrict__ h, const float* __restrict__ W3,
                            float* __restrict__ t, int N, int H) {
  int i = blockIdx.x * blockDim.x + threadIdx.x;
  if (i >= N) return;
  float acc = 0.0f;
  const float* hp = h + (size_t)i * H;
#pragma unroll 8
  for (int c = 0; c < H; ++c) acc += hp[c] * W3[c];
  t[i] = acc;
}
__global__ void k_score_self(const float* __restrict__ t, const float* __restrict__ dis,
                             const float* __restrict__ b3, float* __restrict__ sP, int N) {
  int i = blockIdx.x * blockDim.x + threadIdx.x;
  if (i >= N) return;
  float d = dis[i];
  sP[i] = d * d * t[i] + b3[0];
}
__global__ void k_score_scatter(const int* __restrict__ src, const int* __restrict__ dst,
                                const float* __restrict__ dis, const float* __restrict__ t,
                                float* __restrict__ sP, int E) {
  int e = blockIdx.x * blockDim.x + threadIdx.x;
  if (e >= E) return;
  int s = src[e], d = dst[e];
  atomicAdd(&sP[d], dis[s] * dis[d] * t[s]);
}

// ---------------- top-k: stable LSD radix sort (4-bit digits, 8 passes) ----------------
// key = ~orderable(score) -> ascending radix sort = descending score, LSD
// stability reproduces jax.lax.top_k's index-ascending tie-break.
#define RX_EPT 4
#define RX_EPB (TPB * RX_EPT)   // 1024 elems / block

__global__ void k_key_init(const float* __restrict__ sP, unsigned* keys, unsigned* idx, int n) {
  int i = blockIdx.x * blockDim.x + threadIdx.x;
  if (i >= n) return;
  unsigned u = __float_as_uint(sP[i]);
  unsigned mu = (u & 0x80000000u) ? ~u : (u | 0x80000000u);  // ascending-order map
  keys[i] = ~mu;                                             // descending
  idx[i] = (unsigned)i;
}
__global__ void k_radix_count(const unsigned* __restrict__ keys, unsigned* counts,
                              int n, int shift, int nblocks) {
  __shared__ unsigned hist[16];
  int t = threadIdx.x;
  if (t < 16) hist[t] = 0;
  __syncthreads();
  int base = blockIdx.x * RX_EPB;
#pragma unroll
  for (int j = 0; j < RX_EPT; ++j) {
    int i = base + t * RX_EPT + j;
    if (i < n) atomicAdd(&hist[(keys[i] >> shift) & 15u], 1u);
  }
  __syncthreads();
  if (t < 16) counts[t * nblocks + blockIdx.x] = hist[t];   // digit-major
}
__global__ void k_radix_scan(const unsigned* __restrict__ counts, unsigned* offs, int total) {
  if (blockIdx.x == 0 && threadIdx.x == 0) {
    unsigned run = 0;
    for (int i = 0; i < total; ++i) { offs[i] = run; run += counts[i]; }
  }
}
__global__ void k_radix_scatter(const unsigned* __restrict__ keys, const unsigned* __restrict__ vals,
                                unsigned* __restrict__ dkeys, unsigned* __restrict__ dvals,
                                const unsigned* __restrict__ offs, int n, int shift, int nblocks) {
  __shared__ unsigned hist[TPB * 16];   // per-thread digit histogram, 16 KB
  int t = threadIdx.x;
#pragma unroll
  for (int d = 0; d < 16; ++d) hist[t * 16 + d] = 0;
  int base = blockIdx.x * RX_EPB;
  unsigned kq[RX_EPT], vq[RX_EPT];
  int dg[RX_EPT]; bool ok[RX_EPT];
#pragma unroll
  for (int j = 0; j < RX_EPT; ++j) {
    int i = base + t * RX_EPT + j;
    ok[j] = (i < n);
    if (ok[j]) {
      kq[j] = keys[i]; vq[j] = vals[i];
      dg[j] = (int)((kq[j] >> shift) & 15u);
      hist[t * 16 + dg[j]]++;
    } else dg[j] = 0;
  }
  __syncthreads();
  // exclusive scan per digit across threads (order = original element order)
  if (t < 16) {
    unsigned run = 0;
    for (int i = 0; i < TPB; ++i) {
      unsigned c = hist[i * 16 + t];
      hist[i * 16 + t] = run;
      run += c;
    }
  }
  __syncthreads();
#pragma unroll
  for (int j = 0; j < RX_EPT; ++j) {
    if (!ok[j]) continue;
    unsigned lr = hist[t * 16 + dg[j]];
#pragma unroll
    for (int j2 = 0; j2 < RX_EPT; ++j2)
      if (j2 < j && ok[j2] && dg[j2] == dg[j]) lr++;
    unsigned pos = offs[dg[j] * nblocks + blockIdx.x] + lr;
    dkeys[pos] = kq[j];
    dvals[pos] = vq[j];
  }
}

// ---------------- outputs ----------------
__global__ void k_xnew(const float* __restrict__ x, const float* __restrict__ sP,
                       const unsigned* __restrict__ perm, float* __restrict__ out,
                       int k, int C) {
  int tid = blockIdx.x * blockDim.x + threadIdx.x;
  if (tid >= k * C) return;
  int i = tid >> 7, c = tid & 127;          // C == 128
  int p = (int)perm[i];
  out[tid] = x[(size_t)p * C + c] * tanhf(sP[p]);
}
__global__ void k_newid_init(int* newid, int n) {
  int i = blockIdx.x * blockDim.x + threadIdx.x;
  if (i < n) newid[i] = -1;
}
__global__ void k_newid_set(const unsigned* __restrict__ perm, int* newid, int k) {
  int i = blockIdx.x * blockDim.x + threadIdx.x;
  if (i < k) newid[perm[i]] = i;
}
__global__ void k_edge_out(const int* __restrict__ src, const int* __restrict__ dst,
                           const int* __restrict__ newid, float* __restrict__ out, int E) {
  int e = blockIdx.x * blockDim.x + threadIdx.x;
  if (e >= E) return;
  int ns = newid[src[e]], nd = newid[dst[e]];
  bool keep = (ns >= 0) && (nd >= 0);
  out[e]     = keep ? (float)ns : -1.0f;
  out[E + e] = keep ? (float)nd : -1.0f;
}
__global__ void k_tail_out(const unsigned* __restrict__ perm, float* __restrict__ out, int k) {
  int i = blockIdx.x * blockDim.x + threadIdx.x;
  if (i >= k) return;
  out[i]     = 0.0f;            // batch_new
  out[k + i] = (float)perm[i];  // perm
}

// ---------------- host ----------------
extern "C" void kernel_launch(void* const* d_in, const int* in_sizes, int n_in,
                              void* d_out, int out_size, void* d_ws, size_t ws_size,
                              hipStream_t stream) {
  const float* x  = (const float*)d_in[0];
  const int*   ei = (const int*)d_in[1];
  const float* W1 = (const float*)d_in[2];
  const float* b1 = (const float*)d_in[3];
  const float* W2 = (const float*)d_in[4];
  const float* b2 = (const float*)d_in[5];
  const float* W3 = (const float*)d_in[6];
  const float* b3 = (const float*)d_in[7];

  const int H = in_sizes[3];            // 32
  const int C = in_sizes[2] / H;        // 128
  const int N = in_sizes[0] / C;        // 100000
  const int E = in_sizes[1] / 2;        // 640000
  const int Ktop = (4 * N + 4) / 5;     // ceil(0.8*N) = 80000
  const int* src = ei;
  const int* dst = ei + E;

  char* ws = (char*)d_ws;
  size_t off = 0;
  auto carve = [&](size_t bytes) -> void* {
    void* p = ws + off;
    off += (bytes + 255) & ~(size_t)255;
    return p;
  };
  float* dis  = (float*)carve((size_t)N * 4);
  float* hW   = (float*)carve((size_t)N * H * 4);
  float* h    = (float*)carve((size_t)N * H * 4);
  float* tsc  = (float*)carve((size_t)N * 4);
  float* sP   = (float*)carve((size_t)N * 4);
  unsigned* keysA = (unsigned*)carve((size_t)N * 4);
  unsigned* idxA  = (unsigned*)carve((size_t)N * 4);
  unsigned* keysB = (unsigned*)carve((size_t)N * 4);
  unsigned* idxB  = (unsigned*)carve((size_t)N * 4);
  const int nrb = (N + RX_EPB - 1) / RX_EPB;
  unsigned* counts = (unsigned*)carve((size_t)16 * nrb * 4);
  unsigned* offs   = (unsigned*)carve((size_t)16 * nrb * 4);
  int* newid = (int*)carve((size_t)N * 4);

  auto cdiv = [](int a, int b) { return (a + b - 1) / b; };

  // degree + rsqrt normalization
  k_fill_f32<<<cdiv(N, TPB), TPB, 0, stream>>>(dis, 1.0f, N);
  k_deg_accum<<<cdiv(E, TPB), TPB, 0, stream>>>(dst, dis, E);
  k_rsqrt<<<cdiv(N, TPB), TPB, 0, stream>>>(dis, N);

  // GCN layer 1 (C -> 32)
  k_gemm32_wmma<<<cdiv(N, 128), TPB, 0, stream>>>(x, W1, hW, N, C);
  k_selfterm<<<cdiv(N * 32, TPB), TPB, 0, stream>>>(hW, dis, b1, h, N);
  k_edge_scatter<<<cdiv(E * 8, TPB), TPB, 0, stream>>>(src, dst, dis, hW, h, E);
  k_relu<<<cdiv(N * 32, TPB), TPB, 0, stream>>>(h, N * 32);

  // GCN layer 2 (32 -> 32)
  k_gemm32_wmma<<<cdiv(N, 128), TPB, 0, stream>>>(h, W2, hW, N, H);
  k_selfterm<<<cdiv(N * 32, TPB), TPB, 0, stream>>>(hW, dis, b2, h, N);
  k_edge_scatter<<<cdiv(E * 8, TPB), TPB, 0, stream>>>(src, dst, dis, hW, h, E);
  k_relu<<<cdiv(N * 32, TPB), TPB, 0, stream>>>(h, N * 32);

  // GCN layer 3 scoring (32 -> 1)
  k_score_pre<<<cdiv(N, TPB), TPB, 0, stream>>>(h, W3, tsc, N, H);
  k_score_self<<<cdiv(N, TPB), TPB, 0, stream>>>(tsc, dis, b3, sP, N);
  k_score_scatter<<<cdiv(E, TPB), TPB, 0, stream>>>(src, dst, dis, tsc, sP, E);

  // exact stable descending sort of scores (8x 4-bit LSD radix passes)
  k_key_init<<<cdiv(N, TPB), TPB, 0, stream>>>(sP, keysA, idxA, N);
  unsigned *ka = keysA, *va = idxA, *kb = keysB, *vb = idxB;
  for (int p = 0; p < 8; ++p) {
    int sh = p * 4;
    k_radix_count<<<nrb, TPB, 0, stream>>>(ka, counts, N, sh, nrb);
    k_radix_scan<<<1, 1, 0, stream>>>(counts, offs, 16 * nrb);
    k_radix_scatter<<<nrb, TPB, 0, stream>>>(ka, va, kb, vb, offs, N, sh, nrb);
    unsigned* tk = ka; ka = kb; kb = tk;
    unsigned* tv = va; va = vb; vb = tv;
  }
  // sorted (descending score, index-ascending ties): keys in ka, perm in va

  float* out = (float*)d_out;
  const long kC = (long)Ktop * C;
  k_xnew<<<cdiv(Ktop * C, TPB), TPB, 0, stream>>>(x, sP, va, out, Ktop, C);
  k_newid_init<<<cdiv(N, TPB), TPB, 0, stream>>>(newid, N);
  k_newid_set<<<cdiv(Ktop, TPB), TPB, 0, stream>>>(va, newid, Ktop);
  k_edge_out<<<cdiv(E, TPB), TPB, 0, stream>>>(src, dst, newid, out + kC, E);
  k_tail_out<<<cdiv(Ktop, TPB), TPB, 0, stream>>>(va, out + kC + 2L * E, Ktop);
}